// NWJ_7662221656758
// MI455X (gfx1250) — compile-verified
//
#include <hip/hip_runtime.h>
#include <hip/hip_bf16.h>

typedef float v2f __attribute__((ext_vector_type(2)));
typedef float v8f __attribute__((ext_vector_type(8)));

#define N_S 1024
#define XD  128
#define HH  256
#define PADH 260   // LDS row pitch in floats: mult of 4 (b128 align), ==4 mod 8 -> conflict-free

// ---------------------------------------------------------------------------
// Phase 1: hx = x @ Wx.T ; hyb = y @ Wy.T + b1   via V_WMMA_F32_16X16X4_F32
// One wave per 16x16 output tile. K loop = 128 in steps of 4.
// A-frag (16x4 f32): lanes 0-15 hold K={k,k+1} for M=lane; lanes 16-31 hold
// K={k+2,k+3} for M=lane-16  -> float2 load at (k + 2*half).
// B-frag (4x16 f32): VGPR0 = rows {0,2}, VGPR1 = rows {1,3}, N=lane&15
//   -> symmetric float2 load from W1 row (colbase+l16) at (k + 2*half).
// C/D: VGPR r -> row (8*half + r), col = lane&15.
// ---------------------------------------------------------------------------
__global__ __launch_bounds__(32) void gemm_wmma_kernel(
    const float* __restrict__ x, const float* __restrict__ y,
    const float* __restrict__ W1, const float* __restrict__ b1,
    float* __restrict__ hx, float* __restrict__ hyb)
{
    const int lane = threadIdx.x;
    const int half = lane >> 4;
    const int l16  = lane & 15;
    const int rowbase = blockIdx.x * 16;   // sample rows
    const int colbase = blockIdx.y * 16;   // hidden units
    const int isY = blockIdx.z;

    const float* A    = isY ? y : x;                       // (1024,128)
    const int    koff = isY ? XD : 0;                      // Wy = W1[:,128:]
    const float* arow = A  + (size_t)(rowbase + l16) * XD;
    const float* brow = W1 + (size_t)(colbase + l16) * (2 * XD) + koff;

    v8f acc = {};
    #pragma unroll
    for (int k = 0; k < XD; k += 4) {
        const int ka = k + 2 * half;
        v2f a = *(const v2f*)(arow + ka);
        v2f b = *(const v2f*)(brow + ka);
        acc = __builtin_amdgcn_wmma_f32_16x16x4_f32(
                  false, a, false, b, (short)0, acc, false, false);
    }

    const float bias = isY ? b1[colbase + l16] : 0.0f;
    float* out = isY ? hyb : hx;
    const int mrow = rowbase + 8 * half;
    #pragma unroll
    for (int r = 0; r < 8; ++r)
        out[(size_t)(mrow + r) * HH + colbase + l16] = acc[r] + bias;
}

// ---------------------------------------------------------------------------
// Phase 2: term1[n] = sum_h relu(hx[n,h] + hyb[n,h]) * w2[h] + b2
// ---------------------------------------------------------------------------
__global__ __launch_bounds__(256) void t0_kernel(
    const float* __restrict__ hx, const float* __restrict__ hyb,
    const float* __restrict__ W2, const float* __restrict__ b2,
    float* __restrict__ term1)
{
    const int wave = threadIdx.x >> 5;
    const int lane = threadIdx.x & 31;
    const int row  = blockIdx.x * 8 + wave;

    float s = 0.0f;
    #pragma unroll
    for (int h = lane; h < HH; h += 32)
        s = fmaf(fmaxf(hx[(size_t)row * HH + h] + hyb[(size_t)row * HH + h], 0.0f),
                 W2[h], s);
    #pragma unroll
    for (int m = 16; m >= 1; m >>= 1) s += __shfl_xor(s, m, 32);
    if (lane == 0) term1[row] = s + b2[0];
}

// ---------------------------------------------------------------------------
// Zero the term2 accumulator (ws is poisoned; must re-init every call).
// ---------------------------------------------------------------------------
__global__ void init_kernel(float* __restrict__ term2acc)
{
    term2acc[blockIdx.x * 256 + threadIdx.x] = 0.0f;
}

__device__ __forceinline__ void accum4(float4& acc, const float4 vx,
                                       const float4 vy, const float4 w)
{
    acc.x = fmaf(fmaxf(vx.x + vy.x, 0.0f), w.x, acc.x);
    acc.y = fmaf(fmaxf(vx.y + vy.y, 0.0f), w.y, acc.y);
    acc.z = fmaf(fmaxf(vx.z + vy.z, 0.0f), w.z, acc.z);
    acc.w = fmaf(fmaxf(vx.w + vy.w, 0.0f), w.w, acc.w);
}

// ---------------------------------------------------------------------------
// Phase 3: term2acc[a] += sum_b exp( sum_h relu(hx[b,h]+hyb[a,h])*w2[h] + b2-1 )
// 32x32 (a,b) tile per 256-thread block; 2x2 register tile per thread:
//   a in {a0+ta2, a0+ta2+16}, b in {b0+tb2, b0+tb2+16}
// -> 5 LDS b128 reads per 48 VALU ops (vs 3 per 12 before).
// Dynamic LDS: hxs[32][PADH], hybs[32][PADH], w2s[256]  (67.6 KB).
// ---------------------------------------------------------------------------
__global__ __launch_bounds__(256) void pair_kernel(
    const float* __restrict__ hx, const float* __restrict__ hyb,
    const float* __restrict__ W2, const float* __restrict__ b2,
    float* __restrict__ term2acc)
{
    extern __shared__ float smem[];
    float* hxs  = smem;                 // 32 * PADH
    float* hybs = smem + 32 * PADH;     // 32 * PADH
    float* w2s  = smem + 64 * PADH;     // 256

    const int tid = threadIdx.x;        // 256 threads
    const int b0  = blockIdx.x * 32;
    const int a0  = blockIdx.y * 32;

    // Stage tiles: 32 rows x 256 cols each, float4-coalesced.
    #pragma unroll
    for (int it = 0; it < 8; ++it) {
        const int idx  = it * 256 + tid;      // 0..2047 float4 slots
        const int row  = idx >> 6;            // 64 float4 per row
        const int col4 = idx & 63;
        *(float4*)(hxs  + row * PADH + col4 * 4) =
            *(const float4*)(hx  + (size_t)(b0 + row) * HH + col4 * 4);
        *(float4*)(hybs + row * PADH + col4 * 4) =
            *(const float4*)(hyb + (size_t)(a0 + row) * HH + col4 * 4);
    }
    w2s[tid] = W2[tid];
    __syncthreads();

    const int ta2 = tid >> 4;   // 0..15 -> a rows ta2, ta2+16
    const int tb2 = tid & 15;   // 0..15 -> b rows tb2, tb2+16
    const float4* px0 = (const float4*)(hxs  +  tb2       * PADH);
    const float4* px1 = (const float4*)(hxs  + (tb2 + 16) * PADH);
    const float4* py0 = (const float4*)(hybs +  ta2       * PADH);
    const float4* py1 = (const float4*)(hybs + (ta2 + 16) * PADH);
    const float4* pw  = (const float4*)w2s;

    float4 acc00 = {0,0,0,0}, acc01 = {0,0,0,0};
    float4 acc10 = {0,0,0,0}, acc11 = {0,0,0,0};

    #pragma unroll 4
    for (int h4 = 0; h4 < HH / 4; ++h4) {
        const float4 vx0 = px0[h4];
        const float4 vx1 = px1[h4];
        const float4 vy0 = py0[h4];
        const float4 vy1 = py1[h4];
        const float4 w   = pw[h4];
        accum4(acc00, vx0, vy0, w);   // (a=ta2,    b=tb2)
        accum4(acc01, vx1, vy0, w);   // (a=ta2,    b=tb2+16)
        accum4(acc10, vx0, vy1, w);   // (a=ta2+16, b=tb2)
        accum4(acc11, vx1, vy1, w);   // (a=ta2+16, b=tb2+16)
    }

    const float c1 = b2[0] - 1.0f;
    const float t00 = (acc00.x + acc00.y) + (acc00.z + acc00.w) + c1;
    const float t01 = (acc01.x + acc01.y) + (acc01.z + acc01.w) + c1;
    const float t10 = (acc10.x + acc10.y) + (acc10.z + acc10.w) + c1;
    const float t11 = (acc11.x + acc11.y) + (acc11.z + acc11.w) + c1;

    float ea0 = __expf(t00) + __expf(t01);   // sum over this thread's b for a=a0+ta2
    float ea1 = __expf(t10) + __expf(t11);   // ... for a=a0+ta2+16

    // reduce over tb2 (xor masks < 16 stay within the 16-lane half-wave,
    // where ta2 is constant)
    #pragma unroll
    for (int m = 1; m <= 8; m <<= 1) {
        ea0 += __shfl_xor(ea0, m, 32);
        ea1 += __shfl_xor(ea1, m, 32);
    }
    if (tb2 == 0) {
        atomicAdd(&term2acc[a0 + ta2],      ea0);
        atomicAdd(&term2acc[a0 + ta2 + 16], ea1);
    }
}

// ---------------------------------------------------------------------------
// Phase 4: final scalars.
// ---------------------------------------------------------------------------
__global__ __launch_bounds__(256) void final_kernel(
    const float* __restrict__ term1, const float* __restrict__ term2acc,
    const float* __restrict__ lam, float* __restrict__ out)
{
    const int tid = threadIdx.x;
    float s1 = 0.0f, s2 = 0.0f, q1 = 0.0f, q2 = 0.0f;
    for (int i = tid; i < N_S; i += 256) {
        const float t1 = term1[i];
        const float t2 = term2acc[i] * (1.0f / N_S);
        s1 += t1; s2 += t2;
        q1 = fmaf(t1, t1, q1); q2 = fmaf(t2, t2, q2);
    }
    #pragma unroll
    for (int m = 16; m >= 1; m >>= 1) {
        s1 += __shfl_xor(s1, m, 32); s2 += __shfl_xor(s2, m, 32);
        q1 += __shfl_xor(q1, m, 32); q2 += __shfl_xor(q2, m, 32);
    }
    __shared__ float rs1[8], rs2[8], rq1[8], rq2[8];
    const int wave = tid >> 5, lane = tid & 31;
    if (lane == 0) { rs1[wave] = s1; rs2[wave] = s2; rq1[wave] = q1; rq2[wave] = q2; }
    __syncthreads();
    if (tid == 0) {
        float S1 = 0, S2 = 0, Q1 = 0, Q2 = 0;
        for (int w = 0; w < 8; ++w) { S1 += rs1[w]; S2 += rs2[w]; Q1 += rq1[w]; Q2 += rq2[w]; }
        const float m1 = S1 * (1.0f / N_S), m2 = S2 * (1.0f / N_S);
        const float lb = m1 - m2;
        const float vr = (Q1 * (1.0f / N_S) - m1 * m1) + (Q2 * (1.0f / N_S) - m2 * m2);
        out[0] = lb;
        out[1] = lb - lam[0] * vr;
    }
}

extern "C" void kernel_launch(void* const* d_in, const int* in_sizes, int n_in,
                              void* d_out, int out_size, void* d_ws, size_t ws_size,
                              hipStream_t stream)
{
    const float* x   = (const float*)d_in[0];   // (1024,128)
    const float* y   = (const float*)d_in[1];   // (1024,128)
    const float* lam = (const float*)d_in[2];   // scalar
    const float* W1  = (const float*)d_in[3];   // (256,256)
    const float* b1  = (const float*)d_in[4];   // (256,)
    const float* W2  = (const float*)d_in[5];   // (1,256)
    const float* b2  = (const float*)d_in[6];   // (1,)
    float* out = (float*)d_out;

    float* ws    = (float*)d_ws;
    float* hx    = ws;                          // 1024*256
    float* hyb   = ws + (size_t)N_S * HH;       // 1024*256
    float* term1 = ws + (size_t)2 * N_S * HH;   // 1024
    float* term2 = term1 + N_S;                 // 1024

    dim3 g1(N_S / 16, HH / 16, 2);
    gemm_wmma_kernel<<<g1, 32, 0, stream>>>(x, y, W1, b1, hx, hyb);
    t0_kernel<<<N_S / 8, 256, 0, stream>>>(hx, hyb, W2, b2, term1);
    init_kernel<<<N_S / 256, 256, 0, stream>>>(term2);

    const size_t pair_lds = (size_t)(64 * PADH + HH) * sizeof(float); // 67.6 KB
    pair_kernel<<<dim3(N_S / 32, N_S / 32), 256, pair_lds, stream>>>(
        hx, hyb, W2, b2, term2);

    final_kernel<<<1, 256, 0, stream>>>(term1, term2, lam, out);
}